// AdaptiveGCN_71863392797007
// MI455X (gfx1250) — compile-verified
//
#include <hip/hip_runtime.h>
#include <hip/hip_bf16.h>
#include <stdint.h>

typedef __attribute__((ext_vector_type(2))) float v2f;
typedef __attribute__((ext_vector_type(8))) float v8f;

__device__ __forceinline__ v8f wmma_f32_k4(v2f a, v2f b, v8f c) {
  // V_WMMA_F32_16X16X4_F32 : D = A(16x4) * B(4x16) + C(16x16), fp32 throughout
  return __builtin_amdgcn_wmma_f32_16x16x4_f32(false, a, false, b, (short)0, c,
                                               false, false);
}

// ---------------------------------------------------------------------------
// Kernel 1: adp = softmax(relu(nv1 @ nv2), axis=-1)   (512x10 @ 10x512)
// ---------------------------------------------------------------------------
__global__ __launch_bounds__(256) void adp_kernel(const float* __restrict__ nv1,
                                                  const float* __restrict__ nv2,
                                                  float* __restrict__ adp) {
  __shared__ float red[256];
  const int i = blockIdx.x;     // row
  const int t = threadIdx.x;    // two columns per thread: t, t+256
  float e[10];
#pragma unroll
  for (int k = 0; k < 10; ++k) e[k] = nv1[i * 10 + k];
  float v0 = 0.f, v1 = 0.f;
#pragma unroll
  for (int k = 0; k < 10; ++k) {
    v0 += e[k] * nv2[k * 512 + t];
    v1 += e[k] * nv2[k * 512 + t + 256];
  }
  v0 = fmaxf(v0, 0.f);
  v1 = fmaxf(v1, 0.f);
  red[t] = fmaxf(v0, v1);
  __syncthreads();
  for (int s = 128; s > 0; s >>= 1) {
    if (t < s) red[t] = fmaxf(red[t], red[t + s]);
    __syncthreads();
  }
  const float mx = red[0];
  __syncthreads();
  const float x0 = __expf(v0 - mx), x1 = __expf(v1 - mx);
  red[t] = x0 + x1;
  __syncthreads();
  for (int s = 128; s > 0; s >>= 1) {
    if (t < s) red[t] += red[t + s];
    __syncthreads();
  }
  const float inv = 1.f / red[0];
  adp[i * 512 + t] = x0 * inv;
  adp[i * 512 + t + 256] = x1 * inv;
}

// ---------------------------------------------------------------------------
// Kernel 2: S = A @ A for a 512x512 row-major matrix (tiny: 0.3% of FLOPs)
// ---------------------------------------------------------------------------
__global__ __launch_bounds__(256) void sq_kernel(const float* __restrict__ A,
                                                 float* __restrict__ S) {
  const int w = blockIdx.x >> 1;
  const int v = ((blockIdx.x & 1) << 8) + threadIdx.x;
  float s = 0.f;
#pragma unroll 4
  for (int k = 0; k < 512; ++k) s += A[w * 512 + k] * A[k * 512 + v];
  S[w * 512 + v] = s;
}

// ---------------------------------------------------------------------------
// Main kernel: one block per batch-slice p (512 blocks, 256 threads = 8 waves)
//   Out_p = bias + W0 @ X_p + sum_{i=1..6} Wi @ (X_p @ P_i)
// X_p (128x512 fp32, padded stride 516) staged in LDS via async-to-LDS loads.
// Each wave owns one 16-row M-block and processes TWO 16-wide N-tiles at a
// time (shared A fragments -> half the ds/W traffic per WMMA). The 128x16x2
// intermediate T tiles are exchanged between the two WMMA chains through a
// small shared LDS buffer.
// ---------------------------------------------------------------------------
__global__ __launch_bounds__(256, 1) void gcn_main(
    const float* __restrict__ x,
    const float* __restrict__ P1, const float* __restrict__ P2,
    const float* __restrict__ P3, const float* __restrict__ P4,
    const float* __restrict__ P5, const float* __restrict__ P6,
    const float* __restrict__ W, const float* __restrict__ bias,
    float* __restrict__ out) {
  __shared__ float Xs[128 * 516];     // 264192 B, X_p[c][v], row stride 516
  __shared__ float Ts[2 * 16 * 132];  //  16896 B, T[tile][n][c], stride 132

  const int p  = blockIdx.x;
  const int b0 = p >> 6;
  const int q  = p & 63;
  const int tid = threadIdx.x;

  // ---- Stage X_p into LDS with GLOBAL_LOAD_ASYNC_TO_LDS_B32 (ASYNCcnt) ----
  // xr[p, cc, vv] = x[b0, vv&63, q*8 + (cc>>4), (cc&15)*8 + ((vv>>6)&7)]
  // Iterate in source-contiguous order for perfect global coalescing;
  // per-lane LDS destinations resolve the permutation on the fly.
  for (int it = 0; it < 256; ++it) {
    const int chunk = it * 2 + (tid >> 7);  // 0..511 = (n8, l)
    const int c  = tid & 127;
    const int n8 = chunk >> 6;
    const int l  = chunk & 63;
    const int n  = q * 8 + n8;
    const int cc = n8 * 16 + (c >> 3);
    const int vv = (c & 7) * 64 + l;
    // generic LDS pointers carry the wave-relative LDS offset in bits [31:0]
    const uint32_t laddr = (uint32_t)(uintptr_t)(&Xs[cc * 516 + vv]);
    const uint64_t gaddr =
        (uint64_t)(uintptr_t)(x + (((b0 * 64 + l) * 512 + n) * 128) + c);
    asm volatile("global_load_async_to_lds_b32 %0, %1, off"
                 :
                 : "v"(laddr), "v"(gaddr)
                 : "memory");
  }
  asm volatile("s_wait_asynccnt 0x0" ::: "memory");
  __syncthreads();

  const int lane = tid & 31;
  const int g  = lane & 15;   // column / M-row-within-16
  const int hi = lane >> 4;   // K-half select
  const int m  = tid >> 5;    // wave id == M-block (rows m*16 .. m*16+15)

  // Bias per accumulator row: co = m*16 + r + 8*hi  (tile-invariant)
  float biasreg[8];
#pragma unroll
  for (int r = 0; r < 8; ++r) biasreg[r] = bias[m * 16 + r + 8 * hi];

  const float* Pm[6] = {P1, P2, P3, P4, P5, P6};

  for (int tp = 0; tp < 16; ++tp) {  // tile pair: columns [tp*32, tp*32+32)
    v8f acc0, acc1;
#pragma unroll
    for (int r = 0; r < 8; ++r) { acc0[r] = biasreg[r]; acc1[r] = biasreg[r]; }

    // ---- term i = 0: acc += W0(mblock,:) @ X_p[:, tiles]  (B from Xs) ----
    {
      const float* Wr = W + (m * 16 + g) * 896;  // chunk 0
      const int col0 = tp * 32 + g;
#pragma unroll 4
      for (int k0 = 0; k0 < 128; k0 += 4) {
        v2f A = *(const v2f*)(Wr + k0 + 2 * hi);
        const float* xc = &Xs[(k0 + 2 * hi) * 516 + col0];
        v2f B0, B1;
        B0.x = xc[0];   B0.y = xc[516];
        B1.x = xc[16];  B1.y = xc[516 + 16];
        acc0 = wmma_f32_k4(A, B0, acc0);
        acc1 = wmma_f32_k4(A, B1, acc1);
      }
    }

    for (int i = 0; i < 6; ++i) {
      // ---- matmul1: T(mblock, tiles) = X_p(mblock, :) @ P_i[:, tiles] ----
      // B fragments read P_i rows directly (row-major a / a^2 == P^T layout).
      v8f t0 = {0.f, 0.f, 0.f, 0.f, 0.f, 0.f, 0.f, 0.f};
      v8f t1 = {0.f, 0.f, 0.f, 0.f, 0.f, 0.f, 0.f, 0.f};
      const float* Pw0 = Pm[i] + (tp * 32 + g) * 512;
      const float* Pw1 = Pw0 + 16 * 512;
#pragma unroll 4
      for (int k0 = 0; k0 < 512; k0 += 4) {
        v2f A = *(const v2f*)(&Xs[(m * 16 + g) * 516 + k0 + 2 * hi]);
        t0 = wmma_f32_k4(A, *(const v2f*)(Pw0 + k0 + 2 * hi), t0);
        t1 = wmma_f32_k4(A, *(const v2f*)(Pw1 + k0 + 2 * hi), t1);
      }

      __syncthreads();  // prior matmul2 reads of Ts complete
      // D layout: VGPR r, lane g, half hi -> (M = r + 8*hi, N = g)
#pragma unroll
      for (int r = 0; r < 8; ++r) {
        Ts[g * 132 + m * 16 + r + 8 * hi]        = t0[r];
        Ts[2112 + g * 132 + m * 16 + r + 8 * hi] = t1[r];
      }
      __syncthreads();  // Ts fully written by all 8 waves

      // ---- matmul2: acc += W_i(mblock,:) @ T  (K = 128 channels) ----
      const float* Wr = W + (m * 16 + g) * 896 + (i + 1) * 128;
#pragma unroll 4
      for (int k0 = 0; k0 < 128; k0 += 4) {
        v2f A = *(const v2f*)(Wr + k0 + 2 * hi);
        v2f B0 = *(const v2f*)(&Ts[g * 132 + k0 + 2 * hi]);
        v2f B1 = *(const v2f*)(&Ts[2112 + g * 132 + k0 + 2 * hi]);
        acc0 = wmma_f32_k4(A, B0, acc0);
        acc1 = wmma_f32_k4(A, B1, acc1);
      }
    }

    // ---- store: out flat = o flat: out[p*65536 + co*512 + w] ----
    float* outp = out + p * 65536 + tp * 32 + g;
#pragma unroll
    for (int r = 0; r < 8; ++r) {
      outp[(m * 16 + r + 8 * hi) * 512]      = acc0[r];
      outp[(m * 16 + r + 8 * hi) * 512 + 16] = acc1[r];
    }
  }
}

// ---------------------------------------------------------------------------
extern "C" void kernel_launch(void* const* d_in, const int* in_sizes, int n_in,
                              void* d_out, int out_size, void* d_ws,
                              size_t ws_size, hipStream_t stream) {
  (void)in_sizes; (void)n_in; (void)out_size; (void)ws_size;
  const float* x   = (const float*)d_in[0];
  const float* nv1 = (const float*)d_in[1];
  const float* nv2 = (const float*)d_in[2];
  const float* a1  = (const float*)d_in[3];
  const float* a2  = (const float*)d_in[4];
  const float* w   = (const float*)d_in[5];
  const float* b   = (const float*)d_in[6];
  float* out = (float*)d_out;

  float* wsf = (float*)d_ws;           // needs 4 MiB: adp + 3 squared mats
  float* adp = wsf;                    // 512*512 each
  float* sq1 = wsf + (1 << 18);
  float* sq2 = wsf + (2 << 18);
  float* sqa = wsf + (3 << 18);

  adp_kernel<<<512, 256, 0, stream>>>(nv1, nv2, adp);
  sq_kernel<<<1024, 256, 0, stream>>>(a1, sq1);
  sq_kernel<<<1024, 256, 0, stream>>>(a2, sq2);
  sq_kernel<<<1024, 256, 0, stream>>>(adp, sqa);  // after adp_kernel, same stream

  gcn_main<<<512, 256, 0, stream>>>(x, a1, sq1, a2, sq2, adp, sqa, w, b, out);
}